// AdaptiveThresholding_57904749085171
// MI455X (gfx1250) — compile-verified
//
#include <hip/hip_runtime.h>

// AdaptiveThreshold (mean-C, 11x11, BORDER_REPLICATE) for MI455X / gfx1250.
//
// Memory-bound (~268 MB traffic, ~11.5us floor @ 23.3 TB/s). Pipeline:
//  - interior workgroups: Tensor Data Mover DMA of the 32x76 f32 tile into
//    LDS (tensor_load_to_lds + s_wait_tensorcnt), border workgroups use a
//    clamped cooperative load (BORDER_REPLICATE).
//  - separable 11-tap box filter as two banded matmuls on the f32 WMMA pipe
//    (v_wmma_f32_16x16x4_f32) -> full f32 precision for the threshold.
//  - non-temporal stores for the write-once output.

typedef __attribute__((ext_vector_type(2))) float v2f;
typedef __attribute__((ext_vector_type(8))) float v8f;
typedef __attribute__((ext_vector_type(4))) unsigned int v4u;
typedef __attribute__((ext_vector_type(8))) int v8i;
typedef __attribute__((ext_vector_type(4))) int v4i;

#define IMG_H   512
#define IMG_W   512
#define TILE_H  16
#define TILE_W  64          // 4 waves x 16 cols
#define PAD     5           // blockSize 11 -> halo 5
#define XROWS   32          // 16 rows + halo, rounded to 2 WMMA row-batches
#define XCOLS   76          // 64 cols + halo, rounded to WMMA K chunks
#define XSTRIDE 76          // == XCOLS: LDS tile is contiguous -> TDM-friendly
#define HSTRIDE 72          // 2*72 % 64 = 16 -> halves hit disjoint banks
#define NCHUNK  7           // ceil(26 / 4) K-chunks of the banded matmul

// 1.0f iff n <= j <= n+10 (banded box-filter matrix element, branch-free)
__device__ __forceinline__ float band(int j, int n) {
    return ((unsigned)(j - n) <= 10u) ? 1.0f : 0.0f;
}

__global__ __launch_bounds__(128)
void adaptive_thresh_wmma(const float* __restrict__ x,
                          float* __restrict__ out) {
    __shared__ __align__(16) float Xs[XROWS * XSTRIDE]; // 9728 B input tile
    __shared__ __align__(16) float Hs[XROWS * HSTRIDE]; // 9216 B horiz sums

    const int img = blockIdx.z;
    const int r0  = blockIdx.y * TILE_H;
    const int c0  = blockIdx.x * TILE_W;
    const float* src = x   + (size_t)img * (IMG_H * IMG_W);
    float*       dst = out + (size_t)img * (IMG_H * IMG_W);
    const int tid = threadIdx.x;

    // Does the 32x76 window lie fully inside the image (no replicate needed)?
    const bool interior =
        (blockIdx.y >= 1) && (blockIdx.y <= IMG_H / TILE_H - 2) &&
        (blockIdx.x >= 1) && (blockIdx.x <= IMG_W / TILE_W - 2);

    if (interior) {
        // --- Tensor Data Mover: DMA 76x32 f32 tile -> contiguous LDS ---
        if (tid < 32) {  // one wave issues the DMA (EXEC-independent op)
            unsigned long long ga = (unsigned long long)(const void*)
                (src + (size_t)(r0 - PAD) * IMG_W + (c0 - PAD));
            unsigned int lds = (unsigned int)(unsigned long long)(void*)Xs;

            v4u g0;                       // D# group 0 (ISA 8.3)
            g0.x = 1u;                    // count=1, user descriptor
            g0.y = lds;                   // lds_addr (bytes)
            g0.z = (unsigned)(ga & 0xffffffffull);          // global_addr lo
            g0.w = (unsigned)((ga >> 32) & 0x01ffffffull)   // global_addr hi
                 | (2u << 30);                              // type = image
            v8i g1;                       // D# group 1 (ISA 8.4)
            g1[0] = 0x20000;              // data_size = 4B, no multicast/pad
            g1[1] = (int)((unsigned)IMG_W << 16);  // tensor_dim0 = 512
            g1[2] = 0;                    // dim0 hi | tensor_dim1 lo16 (=0)
            g1[3] = 0x40 | (XCOLS << 16); // tensor_dim1 = 1<<22 | tile_dim0=76
            g1[4] = XROWS;                // tile_dim1 = 32, tile_dim2 = 0
            g1[5] = IMG_W;                // tensor_dim0_stride = 512
            g1[6] = 0;                    // stride hi | dim1_stride lo
            g1[7] = 0;
            v4i g2;                       // D# group 2: dims 2/3 unused
            g2[0] = 1; g2[1] = 1; g2[2] = 0; g2[3] = 0;
            v4i g3;                       // D# group 3: dim 4 unused
            g3[0] = 0; g3[1] = 1 << 16; g3[2] = 0; g3[3] = 0;
            v8i g4 = {};                  // extension group (clang-23 arity)

            __builtin_amdgcn_tensor_load_to_lds(g0, g1, g2, g3, g4, /*cpol=*/0);
            __builtin_amdgcn_s_wait_tensorcnt(0);
        }
    } else {
        // --- border tiles: clamped cooperative load (BORDER_REPLICATE) ---
        for (int idx = tid; idx < XROWS * XCOLS; idx += 128) {
            int r = idx / XCOLS;
            int c = idx - r * XCOLS;
            int gr = min(IMG_H - 1, max(0, r0 - PAD + r));
            int gc = min(IMG_W - 1, max(0, c0 - PAD + c));
            Xs[r * XSTRIDE + c] = src[gr * IMG_W + gc];
        }
    }
    __syncthreads();

    const int wave  = tid >> 5;
    const int lane  = tid & 31;
    const int half  = lane >> 4;   // 0: lanes 0-15, 1: lanes 16-31
    const int l16   = lane & 15;
    const int cbase = wave * 16;   // this wave's column slab in the tile

    // Banded 0/1 matrix values, shared by horizontal-B and vertical-A:
    // lanes 0-15 hold K=j0,j0+1; lanes 16-31 hold K=j0+2,j0+3 (j0 = 4k).
    float bnd0[NCHUNK], bnd1[NCHUNK];
    #pragma unroll
    for (int k = 0; k < NCHUNK; ++k) {
        const int j0 = 4 * k + 2 * half;
        bnd0[k] = band(j0,     l16);
        bnd1[k] = band(j0 + 1, l16);
    }

    // --- Horizontal pass: H(32x16) = X(32x26) * Wh(26x16), Wh banded 0/1 ---
    #pragma unroll
    for (int batch = 0; batch < 2; ++batch) {
        v8f acc = {};
        const int row = batch * 16 + l16;
        const v2f* xrow = (const v2f*)&Xs[row * XSTRIDE];
        #pragma unroll
        for (int k = 0; k < NCHUNK; ++k) {
            const int j0 = 4 * k + 2 * half;        // K index of a.x
            v2f a = xrow[(cbase + j0) >> 1];        // even offset: aligned b64
            v2f b; b.x = bnd0[k]; b.y = bnd1[k];    // Wh[j0..j0+1][n]
            acc = __builtin_amdgcn_wmma_f32_16x16x4_f32(
                false, a, false, b, (short)0, acc, false, false);
        }
        // C/D layout: VGPR i -> row i (lanes 0-15) / row i+8 (lanes 16-31)
        #pragma unroll
        for (int i = 0; i < 8; ++i) {
            const int m = batch * 16 + i + half * 8;
            Hs[m * HSTRIDE + cbase + l16] = acc[i];
        }
    }
    // H slab is wave-private (each wave reads only its own 16 columns):
    // LDS ops are in-order per wave, no workgroup barrier needed here.

    // --- Vertical pass: Out(16x16) = Wv(16x26) * H(26x16), Wv banded 0/1 ---
    v8f acc = {};
    #pragma unroll
    for (int k = 0; k < NCHUNK; ++k) {
        const int j0 = 4 * k + 2 * half;
        v2f a; a.x = bnd0[k]; a.y = bnd1[k];        // Wv[M=l16][j0..j0+1]
        v2f b;                                      // H rows j0, j0+1
        b.x = Hs[(j0    ) * HSTRIDE + cbase + l16];
        b.y = Hs[(j0 + 1) * HSTRIDE + cbase + l16];
        acc = __builtin_amdgcn_wmma_f32_16x16x4_f32(
            false, a, false, b, (short)0, acc, false, false);
    }

    // --- Epilogue: threshold compare against LDS-resident source pixel ---
    const float invk = 1.0f / 121.0f;  // 11*11 box mean
    #pragma unroll
    for (int i = 0; i < 8; ++i) {
        const int m     = i + half * 8;            // row within output tile
        const int col_l = cbase + l16;             // col within output tile
        const float xv  = Xs[(m + PAD) * XSTRIDE + col_l + PAD];
        const float t   = acc[i] * invk - 2.0f;
        const float o   = (xv > t) ? 0.0f : 255.0f;
        __builtin_nontemporal_store(o, &dst[(size_t)(r0 + m) * IMG_W + c0 + col_l]);
    }
}

extern "C" void kernel_launch(void* const* d_in, const int* in_sizes, int n_in,
                              void* d_out, int out_size, void* d_ws, size_t ws_size,
                              hipStream_t stream) {
    const float* x = (const float*)d_in[0];
    float* out = (float*)d_out;
    const int n_img = in_sizes[0] / (IMG_H * IMG_W);   // 128
    dim3 grid(IMG_W / TILE_W, IMG_H / TILE_H, n_img);  // 8 x 32 x 128
    dim3 block(128);
    adaptive_thresh_wmma<<<grid, block, 0, stream>>>(x, out);
}